// adaptSplit_dotsim_81312320848588
// MI455X (gfx1250) — compile-verified
//
#include <hip/hip_runtime.h>
#include <hip/hip_bf16.h>

// ---------------------------------------------------------------------------
// adaptSplit_dotsim for MI455X (gfx1250)
//
// score kernel: fused 2x2 avg-pool + 16x16 frame Gram via V_WMMA_F32_16X16X4_F32
//               (T=16 frames == WMMA M=N=16; Gram => SRC0 == SRC1 registers)
// gather kernel: bandwidth-bound float4 permutation copy (dominant cost)
// ---------------------------------------------------------------------------

typedef __attribute__((ext_vector_type(2))) float v2f;
typedef __attribute__((ext_vector_type(8))) float v8f;

#define NB    64      // batches
#define TT    16      // frames
#define CC    512     // channels
#define HWP   196     // 14*14 pixels per frame
#define DTOT  25088   // 49 * 512 pooled feature length
#define WAVES 8
#define KSLICE (DTOT / WAVES)   // 3136, multiple of 4

// ------------------------- scoring / frame selection ------------------------
__global__ __launch_bounds__(256)
void adaptsplit_score_select(const float* __restrict__ x, int* __restrict__ idx_out)
{
    const int b    = blockIdx.x;          // one workgroup per batch
    const int tid  = threadIdx.x;
    const int wave = tid >> 5;
    const int lane = tid & 31;
    const int kh   = lane >> 4;           // K half select (0: K=0,1  1: K=2,3)
    const int m    = lane & 15;           // A row (frame t) == B col (frame s)

    __shared__ float lds_sim[WAVES][256]; // per-wave partial 16x16 Gram
    __shared__ float lds_tot[256];        // reduced Gram
    __shared__ float lds_score[16];

    const float* xb = x + (size_t)b * CC * TT * HWP;

    v8f acc = {0.f, 0.f, 0.f, 0.f, 0.f, 0.f, 0.f, 0.f};

    const int k_begin = wave * KSLICE;
    const int k_end   = k_begin + KSLICE;

    for (int k0 = k_begin; k0 < k_end; k0 += 4) {
        // This lane supplies pooled features xi[m][k0 + 2*kh + r], r=0,1
        v2f a;
#pragma unroll
        for (int r = 0; r < 2; ++r) {
            const int k  = k0 + 2 * kh + r;
            const int c  = k / 49;              // channel
            const int p  = k - c * 49;          // pooled pixel 0..48
            const int ph = p / 7;
            const int pw = p - ph * 7;
            const int base = 28 * ph + 2 * pw;  // (2*ph)*14 + 2*pw
            const float* src = xb + ((size_t)c * TT + m) * HWP;
            // 2x2 average pool, two 8-byte loads (even element offsets)
            float2 u = *(const float2*)(src + base);
            float2 v = *(const float2*)(src + base + 14);
            a[r] = 0.25f * (u.x + u.y + v.x + v.y);
        }
        // Gram: D = A x A^T  (B operand layout == A operand layout here)
        acc = __builtin_amdgcn_wmma_f32_16x16x4_f32(
                  /*neg_a=*/false, a, /*neg_b=*/false, a,
                  /*c_mod=*/(short)0, acc, /*reuse_a=*/false, /*reuse_b=*/false);
    }

    // D layout: VGPR r -> sim[M = r + 8*kh][N = m]
#pragma unroll
    for (int r = 0; r < 8; ++r) {
        const int M = r + 8 * kh;
        lds_sim[wave][M * 16 + m] = acc[r];
    }
    __syncthreads();

    // reduce the 8 wave-partials (256 threads, one Gram entry each)
    float tot = 0.f;
#pragma unroll
    for (int w = 0; w < WAVES; ++w) tot += lds_sim[w][tid];
    lds_tot[tid] = tot;
    __syncthreads();

    // score[t] = (1/(C*T)) * sum_s Gram[t][s] + prior[t]
    if (tid < 16) {
        float s = 0.f;
#pragma unroll
        for (int n = 0; n < 16; ++n) s += lds_tot[tid * 16 + n];
        s *= (1.0f / (float)(CC * TT));          // 1/8192
        s += ((tid & 1) == 0) ? 1.0f : 0.0f;     // prior [1,0,1,0,...]
        lds_score[tid] = s;
    }
    __syncthreads();

    // rank frames; top-8 -> approx (ascending), bottom-8 -> detail (ascending)
    if (tid == 0) {
        int na = 0, nd = 0;
        for (int t = 0; t < 16; ++t) {
            const float st = lds_score[t];
            int rank = 0;
            for (int s = 0; s < 16; ++s) {
                const float ss = lds_score[s];
                if (ss > st || (ss == st && s < t)) ++rank;   // top_k tie-break
            }
            if (rank < 8) idx_out[b * 16 + (na++)]     = t;
            else          idx_out[b * 16 + 8 + (nd++)] = t;
        }
    }
}

// ------------------------------ frame gather --------------------------------
// out[0      .. half) = x_a : out[b][c][j][hw] = x[b][c][approx[b][j]][hw]
// out[half .. 2*half) = x_d : same with detail indices
#define HALF4 12845056LL   // 64*512*8*49 float4 elements per section

__global__ __launch_bounds__(256)
void adaptsplit_gather(const float4* __restrict__ x,
                       const int* __restrict__ idx,
                       float4* __restrict__ out)
{
    const long long i = (long long)blockIdx.x * 256 + threadIdx.x; // < 2*HALF4
    const int sec = (i >= HALF4) ? 1 : 0;
    long long r = i - (long long)sec * HALF4;

    const int b  = (int)(r / 200704);            // 512*8*49
    const int r2 = (int)(r - (long long)b * 200704);
    const int c  = r2 / 392;                     // 8*49
    const int r3 = r2 - c * 392;
    const int j  = r3 / 49;
    const int q  = r3 - j * 49;

    const int t = idx[b * 16 + sec * 8 + j];

    const long long src = (((long long)b * CC + c) * TT + t) * 49 + q;
    const long long dst = (long long)sec * HALF4 +
                          (((long long)b * CC + c) * 8 + j) * 49 + q;
    out[dst] = x[src];
}

// ------------------------------- launcher -----------------------------------
extern "C" void kernel_launch(void* const* d_in, const int* in_sizes, int n_in,
                              void* d_out, int out_size, void* d_ws, size_t ws_size,
                              hipStream_t stream)
{
    const float* x  = (const float*)d_in[0];
    int* idx        = (int*)d_ws;                 // 64*16 ints = 4 KB scratch
    float* out      = (float*)d_out;

    adaptsplit_score_select<<<NB, 256, 0, stream>>>(x, idx);

    const long long total4 = 2 * HALF4;           // 25,690,112 float4 copies
    const int grid = (int)(total4 / 256);         // 100,352 blocks exact
    adaptsplit_gather<<<grid, 256, 0, stream>>>((const float4*)x, idx, (float4*)out);
}